// LocalRegionLevelLoss_28484223107534
// MI455X (gfx1250) — compile-verified
//
#include <hip/hip_runtime.h>
#include <hip/hip_bf16.h>

// ---------------------------------------------------------------------------
// LocalRegionLevelLoss on MI455X (gfx1250, wave32)
//   sims     = positives @ img^T          (B,P,R)
//   neg_sims = negatives @ img^T          (B,P,NNEG,R)
//   loss     = sum softmax_r(sims) * relu(margin + max_n(neg_sims) - sims)
//
// One workgroup per (b,p). 3 waves, each owns one 16-row M-tile of
// X = [positive ; 32 negatives ; pad], producing S (48x48, useful 33x36)
// via V_WMMA_F32_16X16X4_F32 against img_feats[b] staged in LDS by the
// Tensor Data Mover (double buffered, K-chunks of 128).
// ---------------------------------------------------------------------------

#define B_    128
#define P_    20
#define NNEG_ 32
#define R_    36
#define D_    1024
#define MARGIN_ 0.2f

#define KC    128                 // K chunk (elements)
#define NCH   (D_ / KC)           // 8 chunks
#define LROW  (KC + 4)            // padded LDS row stride in dwords (TDM pad)
#define SROW  52                  // padded S row stride in dwords

typedef __attribute__((ext_vector_type(2)))  float v2f;
typedef __attribute__((ext_vector_type(8)))  float v8f;
typedef __attribute__((ext_vector_type(4)))  unsigned int v4u;
typedef __attribute__((ext_vector_type(8)))  int v8i;
typedef __attribute__((ext_vector_type(4)))  int v4i;

// Issue a TDM load of a 36 x 128 f32 tile (rows = R, cols = K-chunk) from a
// row-major 36x1024 tensor into LDS, inserting 4 dwords of pad per 128-dword
// row so the LDS row stride becomes LROW dwords.  D# layout per CDNA5 ISA
// sec 8.3-8.6 (group0: count/lds_addr/global_addr/type; group1: data_size,
// pad, dims, tile dims, strides; groups 2/3 zero => 2D tensor).
// Toolchain: amdgpu-toolchain (clang-23) 6-arg form of the builtin.
__device__ __forceinline__ void tdm_load_img(unsigned ldsAddr,
                                             unsigned long long gaddr) {
  v4u g0;
  g0[0] = 1u;                                              // count=1 (valid)
  g0[1] = ldsAddr;                                         // lds_addr (bytes)
  g0[2] = (unsigned)(gaddr & 0xFFFFFFFFull);               // global_addr lo
  g0[3] = (unsigned)((gaddr >> 32) & 0x01FFFFFFull)        // global_addr[56:32]
        | (2u << 30);                                      // type = 2 (image)
  v8i g1;
  g1[0] = (int)((2u << 16)      // data_size = 2 (4 bytes)
              | (1u << 20)      // pad_enable
              | (6u << 22)      // pad_interval: every 128 dwords
              | (3u << 25));    // pad_amount: 4 dwords
  g1[1] = (int)(((unsigned)D_ & 0xFFFFu) << 16);           // tensor_dim0 lo
  g1[2] = (int)((((unsigned)D_ >> 16) & 0xFFFFu)           // tensor_dim0 hi
              | (((unsigned)R_ & 0xFFFFu) << 16));         // tensor_dim1 lo
  g1[3] = (int)(((unsigned)KC & 0xFFFFu) << 16);           // dim1 hi=0 | tile_dim0
  g1[4] = (int)((unsigned)R_ & 0xFFFFu);                   // tile_dim1 | tile_dim2=0
  g1[5] = (int)(unsigned)D_;                               // tensor_dim0_stride lo
  g1[6] = 0;                                               // stride0 hi | stride1 lo
  g1[7] = 0;
  v4i z4 = {0, 0, 0, 0};
  v8i z8 = {0, 0, 0, 0, 0, 0, 0, 0};
  __builtin_amdgcn_tensor_load_to_lds(g0, g1, z4, z4, z8, /*cpol=*/0);
}

__global__ void __launch_bounds__(96)
llr_loss_kernel(const float* __restrict__ img,   // (B,R,D)
                const float* __restrict__ pos,   // (B,P,D)
                const float* __restrict__ neg,   // (B,P,NNEG,D)
                float* __restrict__ out) {
  __shared__ float shb[2][48 * LROW];   // TDM-staged img chunks (double buf)
  __shared__ float sS[48 * SROW];       // S tile spill
  __shared__ float sE[40];              // sims per region
  __shared__ float sH[40];              // hinge term per region

  const int blk  = blockIdx.x;
  const int b    = blk / P_;
  const int p    = blk - b * P_;
  const int tid  = threadIdx.x;
  const int lane = tid & 31;
  const int wv   = tid >> 5;            // M-tile index 0..2

  // ---- per-lane A pointer: row m = wv*16 + lane%16 of X = [pos; neg; pad]
  const int mrow = (wv << 4) + (lane & 15);
  const float* xrow;
  if (mrow == 0) {
    xrow = pos + ((size_t)b * P_ + p) * (size_t)D_;
  } else {
    int n = mrow - 1;
    if (n > NNEG_ - 1) n = NNEG_ - 1;                      // clamp pad rows
    xrow = neg + (((size_t)b * P_ + p) * (size_t)NNEG_ + n) * (size_t)D_;
  }
  const float* aptr = xrow + ((lane >> 4) << 1);           // + k-sublane offset

  v8f c0 = {}, c1 = {}, c2 = {};

  const unsigned long long gbase =
      (unsigned long long)(size_t)(img + (size_t)b * R_ * D_);
  const unsigned lds0 = (unsigned)(size_t)(const void*)&shb[0][0];
  const unsigned lds1 = (unsigned)(size_t)(const void*)&shb[1][0];

  if (wv == 0) tdm_load_img(lds0, gbase);                  // prefetch chunk 0

  for (int ch = 0; ch < NCH; ++ch) {
    if (wv == 0) {
      if (ch + 1 < NCH) {
        tdm_load_img(((ch + 1) & 1) ? lds1 : lds0,
                     gbase + (unsigned long long)(ch + 1) * KC * sizeof(float));
        __builtin_amdgcn_s_wait_tensorcnt(1);              // chunk ch landed
      } else {
        __builtin_amdgcn_s_wait_tensorcnt(0);
      }
    }
    __syncthreads();                                       // chunk visible

    const float* bbuf = &shb[ch & 1][0];
    const int nrow = lane & 15;
    const int ksub = (lane >> 4) << 1;

#pragma unroll 4
    for (int ks = 0; ks < KC / 4; ++ks) {
      // A fragment: X[m][k], streaming (read-once) -> non-temporal
      v2f a = __builtin_nontemporal_load((const v2f*)aptr);
      aptr += 4;
      // B fragments: B[k][n] = img[n][k] from padded LDS
      const int koff = (ks << 2) + ksub;
      v2f b0 = *(const v2f*)&bbuf[(nrow)      * LROW + koff];
      v2f b1 = *(const v2f*)&bbuf[(nrow + 16) * LROW + koff];
      v2f b2 = *(const v2f*)&bbuf[(nrow + 32) * LROW + koff];
      c0 = __builtin_amdgcn_wmma_f32_16x16x4_f32(false, a, false, b0,
                                                 (short)0, c0, false, false);
      c1 = __builtin_amdgcn_wmma_f32_16x16x4_f32(false, a, false, b1,
                                                 (short)0, c1, false, false);
      c2 = __builtin_amdgcn_wmma_f32_16x16x4_f32(false, a, false, b2,
                                                 (short)0, c2, false, false);
    }
    __syncthreads();                                       // done reading buf
  }

  // ---- spill S tiles: C layout VGPR i -> (M = i | i+8, N = lane%16)
  {
    const int rbase = (wv << 4) + ((lane >> 4) << 3);
    const int col   = lane & 15;
#pragma unroll
    for (int i = 0; i < 8; ++i) {
      sS[(rbase + i) * SROW + col]      = c0[i];
      sS[(rbase + i) * SROW + 16 + col] = c1[i];
      sS[(rbase + i) * SROW + 32 + col] = c2[i];
    }
  }
  __syncthreads();

  // ---- epilogue: per-region hinge term, then softmax-weighted sum
  if (tid < R_) {
    const int r = tid;
    float s = sS[r];                                       // row 0 = positive
    float mxn = -3.0e38f;
#pragma unroll
    for (int n = 0; n < NNEG_; ++n)
      mxn = fmaxf(mxn, sS[(1 + n) * SROW + r]);
    sE[r] = s;
    sH[r] = fmaxf(MARGIN_ + mxn - s, 0.0f);
  }
  __syncthreads();

  if (tid == 0) {
    float mx = -3.0e38f;
    for (int r = 0; r < R_; ++r) mx = fmaxf(mx, sE[r]);
    float den = 0.0f;
    for (int r = 0; r < R_; ++r) den += expf(sE[r] - mx);
    float loss = 0.0f;
    for (int r = 0; r < R_; ++r)
      loss += (expf(sE[r] - mx) / den) * sH[r];
    atomicAdd(out, loss);
  }
}

__global__ void llr_zero_kernel(float* __restrict__ out) { out[0] = 0.0f; }

extern "C" void kernel_launch(void* const* d_in, const int* in_sizes, int n_in,
                              void* d_out, int out_size, void* d_ws,
                              size_t ws_size, hipStream_t stream) {
  (void)in_sizes; (void)n_in; (void)out_size; (void)d_ws; (void)ws_size;
  const float* img = (const float*)d_in[0];   // (B,R,D)
  const float* pos = (const float*)d_in[1];   // (B,P,D)
  const float* neg = (const float*)d_in[2];   // (B,P,NNEG,D)
  float* out = (float*)d_out;

  llr_zero_kernel<<<1, 1, 0, stream>>>(out);
  llr_loss_kernel<<<B_ * P_, 96, 0, stream>>>(img, pos, neg, out);
}